// ASTP_17343077941372
// MI455X (gfx1250) — compile-verified
//
#include <hip/hip_runtime.h>
#include <math.h>

#define C_DIM 128
#define S_DIM 4096
#define H_DIM 16
#define W_DIM 11
#define HWN   (H_DIM * W_DIM)   // 176
#define B_SEG 32
#define O_DIM 256
#define EPS_GEM 1e-6f

typedef __attribute__((ext_vector_type(2))) float v2f;
typedef __attribute__((ext_vector_type(8))) float v8f;

// ---------------------------------------------------------------------------
// Kernel 1: ragged segment-max over S. One block per (channel c, segment b).
// x layout: [C, S, H*W]; threads cover the 176 contiguous hw elements so each
// wave32 issues fully-coalesced 128B global loads. Single pass over the 369MB
// input -> HBM-bound optimum. Prefetch ahead to keep loads in flight.
// ---------------------------------------------------------------------------
__global__ void segmax_kernel(const float* __restrict__ x,
                              const int* __restrict__ seqL,
                              float* __restrict__ pooled /* [B][C][HWN] */) {
    const int c  = blockIdx.x;
    const int b  = blockIdx.y;
    const int hw = threadIdx.x;

    int start = 0;
    for (int i = 0; i < b; ++i) start += seqL[i];   // uniform scalar loads
    const int len = seqL[b];

    if (hw >= HWN) return;

    const float* xp = x + ((size_t)c * S_DIM + start) * HWN + hw;

    float m = -__builtin_inff();
    int s = 0;
    for (; s + 4 <= len; s += 4) {
        __builtin_prefetch(xp + (size_t)(s + 16) * HWN, 0, 0);  // global_prefetch_b8
        float a0 = xp[(size_t)(s + 0) * HWN];
        float a1 = xp[(size_t)(s + 1) * HWN];
        float a2 = xp[(size_t)(s + 2) * HWN];
        float a3 = xp[(size_t)(s + 3) * HWN];
        m = fmaxf(m, fmaxf(fmaxf(a0, a1), fmaxf(a2, a3)));
    }
    for (; s < len; ++s) m = fmaxf(m, xp[(size_t)s * HWN]);

    pooled[((size_t)b * C_DIM + c) * HWN + hw] = m;
}

// ---------------------------------------------------------------------------
// Kernel 2: fp32 WMMA GEMM.  y[n, m] = sum_k  B[k, n] * A[m, k]
//   A = W  [O=256, C=128]  row-major          (M x K)
//   B = pooled viewed as   [K=C, N=B*HWN]     (col n = (b, hw))
// One wave (32 threads) per 16x16 output tile; K swept in steps of 4 with
// v_wmma_f32_16x16x4_f32. All tile dims divide exactly -> no divergence,
// EXEC stays all-ones as the WMMA spec requires.
// Fragment layouts per CDNA5 ISA 7.12.2:
//   A v2f: lanes 0-15 -> M=m0+lane, K = k0+{0,1}; lanes 16-31 -> K = k0+{2,3}
//   B v2f: lanes 0-15 -> N=n0+lane, K = k0+{0,1}; lanes 16-31 -> K = k0+{2,3}
//   C/D v8f: VGPR r -> row m0+r (lanes 0-15) / m0+8+r (lanes 16-31), col = n0+(lane&15)
// ---------------------------------------------------------------------------
__global__ void gemm_wmma_kernel(const float* __restrict__ pooled,
                                 const float* __restrict__ Wmat,
                                 float* __restrict__ y /* [B][O][HWN] */) {
    const int lane  = threadIdx.x;            // one wave per block
    const int n0    = blockIdx.x * 16;        // N tile (over B*HWN = 5632)
    const int m0    = blockIdx.y * 16;        // M tile (over O = 256)
    const int l15   = lane & 15;
    const int khalf = (lane >> 4) * 2;        // 0 for lanes 0-15, 2 for 16-31

    const int n  = n0 + l15;
    const int bb = n / HWN;
    const int hw = n - bb * HWN;

    const float* bcol = pooled + (size_t)bb * C_DIM * HWN + hw;  // stride HWN over k
    const float* arow = Wmat + (size_t)(m0 + l15) * C_DIM;       // stride 1 over k

    v8f acc = {0.f, 0.f, 0.f, 0.f, 0.f, 0.f, 0.f, 0.f};

#pragma unroll 4
    for (int k0 = 0; k0 < C_DIM; k0 += 4) {
        v2f a, bf;
        a.x  = arow[k0 + khalf];
        a.y  = arow[k0 + khalf + 1];
        bf.x = bcol[(size_t)(k0 + khalf) * HWN];
        bf.y = bcol[(size_t)(k0 + khalf + 1) * HWN];
        acc = __builtin_amdgcn_wmma_f32_16x16x4_f32(
            /*neg_a=*/false, a, /*neg_b=*/false, bf,
            /*c_mod=*/(short)0, acc, /*reuse_a=*/false, /*reuse_b=*/false);
    }

    const int mbase = m0 + (lane >> 4) * 8;
    float* yout = y + ((size_t)bb * O_DIM) * HWN + hw;
#pragma unroll
    for (int r = 0; r < 8; ++r) {
        yout[(size_t)(mbase + r) * HWN] = acc[r];
    }
}

// ---------------------------------------------------------------------------
// Kernel 3: GeM pooling. One block per (b, o); 176 threads read the row,
// 4 per-part power sums accumulated in LDS, parts of 4 rows x 11 cols = 44.
// ---------------------------------------------------------------------------
__global__ void gem_pool_kernel(const float* __restrict__ y,
                                const float* __restrict__ p,
                                float* __restrict__ out /* [B*O][4] */) {
    const int bo = blockIdx.x;
    const int t  = threadIdx.x;

    __shared__ float parts[4];
    if (t < 4) parts[t] = 0.f;
    __syncthreads();

    if (t < HWN) {
        const int h    = t / W_DIM;
        const int part = h >> 2;                 // SPLIT_PARAM = [4,4,4,4]
        float v = y[(size_t)bo * HWN + t];
        v = fmaxf(v, EPS_GEM);
        atomicAdd(&parts[part], __powf(v, p[part]));
    }
    __syncthreads();

    if (t < 4) {
        out[(size_t)bo * 4 + t] = __powf(parts[t] * (1.0f / 44.0f), 1.0f / p[t]);
    }
}

// ---------------------------------------------------------------------------
extern "C" void kernel_launch(void* const* d_in, const int* in_sizes, int n_in,
                              void* d_out, int out_size, void* d_ws, size_t ws_size,
                              hipStream_t stream) {
    const float* x    = (const float*)d_in[0];
    const int*   seqL = (const int*)d_in[1];
    const float* Wm   = (const float*)d_in[2];
    const float* p    = (const float*)d_in[3];

    float* pooled = (float*)d_ws;                                   // 32*128*176 f32 = 2.88 MB
    float* y      = pooled + (size_t)B_SEG * C_DIM * HWN;           // 32*256*176 f32 = 5.77 MB
    float* out    = (float*)d_out;

    // 1) ragged segment max (single streaming pass over x)
    segmax_kernel<<<dim3(C_DIM, B_SEG), 192, 0, stream>>>(x, seqL, pooled);

    // 2) fp32 WMMA GEMM: M=256, N=5632, K=128 (exact 16x16x4 tiling)
    gemm_wmma_kernel<<<dim3((B_SEG * HWN) / 16, O_DIM / 16), 32, 0, stream>>>(pooled, Wm, y);

    // 3) GeM pooling to [B, O, 4]
    gem_pool_kernel<<<dim3(B_SEG * O_DIM), 192, 0, stream>>>(y, p, out);
}